// Attention_1116691497739
// MI455X (gfx1250) — compile-verified
//
#include <hip/hip_runtime.h>
#include <cmath>

typedef __attribute__((ext_vector_type(2))) float v2f;
typedef __attribute__((ext_vector_type(8))) float v8f;

#define B_ 64
#define T_ 2048
#define Q_ 512
#define K_ 512

// ---------------------------------------------------------------------------
// Kernel 1: mids[b,k] = sum_q W[k,q] * query[b,q]   (tiny: 33 MFLOP, 1 MB read)
// One wave per 16x16 output tile of [B_, K_]; K-dim = Q_, stepped by 4.
// A tile = query rows (16 b's x 4 q's), B tile = W rows read as columns.
// ---------------------------------------------------------------------------
__global__ __launch_bounds__(128) void mids_kernel(const float* __restrict__ query,
                                                   const float* __restrict__ W,
                                                   float* __restrict__ mids) {
    const int wid  = blockIdx.x * (blockDim.x >> 5) + (threadIdx.x >> 5);
    const int lane = threadIdx.x & 31;
    const int half = lane >> 4;   // 0: K=0,1   1: K=2,3
    const int l16  = lane & 15;
    const int tm   = wid >> 5;    // 0..3   (64/16 B-tiles)
    const int tn   = wid & 31;    // 0..31  (512/16 K-tiles)

    const float* qrow = query + (size_t)(tm * 16 + l16) * Q_;  // A: M row = l16
    const float* wrow = W     + (size_t)(tn * 16 + l16) * Q_;  // B: N col = l16 (W[k,:])

    v8f acc = {};
    for (int q0 = 0; q0 < Q_; q0 += 4) {
        v2f a = *(const v2f*)(qrow + q0 + half * 2);
        v2f b = *(const v2f*)(wrow + q0 + half * 2);
        acc = __builtin_amdgcn_wmma_f32_16x16x4_f32(false, a, false, b,
                                                    (short)0, acc, false, false);
    }
    // D: VGPR r holds M=r (lanes 0-15) / M=8+r (lanes 16-31); N = lane&15
    #pragma unroll
    for (int r = 0; r < 8; ++r) {
        const int m = half * 8 + r;
        mids[(size_t)(tm * 16 + m) * K_ + tn * 16 + l16] = acc[r];
    }
}

// ---------------------------------------------------------------------------
// Kernel 2: out[b,t] = exp(tanh(key[b,t,:] . mids[b,:] + bias))   (HBM-bound)
// One wave per (b, 16-t tile): streams 16x512 f32 of key through
// v_wmma_f32_16x16x4_f32 with the mids chunk broadcast across all N columns.
// 8192 waves total -> plenty of latency hiding for the 23.3 TB/s stream.
// ---------------------------------------------------------------------------
__global__ __launch_bounds__(256) void scores_kernel(const float* __restrict__ key,
                                                     const float* __restrict__ mids,
                                                     const float* __restrict__ bias,
                                                     float* __restrict__ out) {
    const int wid  = blockIdx.x * (blockDim.x >> 5) + (threadIdx.x >> 5); // 0..8191
    const int lane = threadIdx.x & 31;
    const int half = lane >> 4;
    const int l16  = lane & 15;
    const int b    = wid >> 7;       // 0..63
    const int t0   = (wid & 127) * 16;

    const float* keyr = key  + ((size_t)b * T_ + t0 + l16) * K_;  // this lane's M row
    const float* midb = mids + (size_t)b * K_;

    v8f acc = {};
    for (int k0 = 0; k0 < K_; k0 += 32) {
        // speculative lookahead for the streaming row (dropped silently if OOB)
        __builtin_prefetch(keyr + k0 + 128, 0, 0);
        #pragma unroll
        for (int kk = 0; kk < 32; kk += 4) {
            v2f a  = *(const v2f*)(keyr + k0 + kk + half * 2);           // A: key tile
            v2f bm = *(const v2f*)(midb + k0 + kk + half * 2);           // B: broadcast
            acc = __builtin_amdgcn_wmma_f32_16x16x4_f32(false, a, false, bm,
                                                        (short)0, acc, false, false);
        }
    }

    // Every column of D is identical (B columns equal), so each lane holds the
    // 8 dot products for rows M = half*8 + r. tanh in (-1,1) => no softmax max.
    const float bv = bias[0];
    float e[8];
    #pragma unroll
    for (int r = 0; r < 8; ++r)
        e[r] = __expf(tanhf(acc[r] + bv));

    if (l16 == 0) {  // lanes 0 and 16 cover t0..t0+7 and t0+8..t0+15
        float* op = out + (size_t)b * T_ + t0 + half * 8;
        #pragma unroll
        for (int r = 0; r < 8; ++r) op[r] = e[r];
    }
}

// ---------------------------------------------------------------------------
// Kernel 3: deterministic per-row normalize: out[b,t] /= sum_t out[b,t]
// One 256-thread block per b; 8 values/thread; LDS tree reduction.
// ---------------------------------------------------------------------------
__global__ __launch_bounds__(256) void softmax_norm_kernel(float* __restrict__ out) {
    __shared__ float sdata[256];
    const int b   = blockIdx.x;
    const int tid = threadIdx.x;
    float* row = out + (size_t)b * T_;

    float v[8];
    float s = 0.f;
    #pragma unroll
    for (int i = 0; i < 8; ++i) {
        v[i] = row[tid + i * 256];
        s += v[i];
    }
    sdata[tid] = s;
    __syncthreads();
    for (int off = 128; off > 0; off >>= 1) {
        if (tid < off) sdata[tid] += sdata[tid + off];
        __syncthreads();
    }
    const float inv = 1.0f / sdata[0];
    #pragma unroll
    for (int i = 0; i < 8; ++i) row[tid + i * 256] = v[i] * inv;
}

// ---------------------------------------------------------------------------
extern "C" void kernel_launch(void* const* d_in, const int* in_sizes, int n_in,
                              void* d_out, int out_size, void* d_ws, size_t ws_size,
                              hipStream_t stream) {
    const float* query = (const float*)d_in[0];  // [64, 512]
    const float* key   = (const float*)d_in[1];  // [64, 2048, 512]
    const float* W     = (const float*)d_in[2];  // [512, 512]
    const float* bias  = (const float*)d_in[3];  // [1]
    float* out  = (float*)d_out;                 // [64, 2048]
    float* mids = (float*)d_ws;                  // [64, 512] = 128 KB scratch

    // 128 tiles -> 32 blocks x 4 waves
    mids_kernel<<<32, 128, 0, stream>>>(query, W, mids);
    // 8192 waves -> 1024 blocks x 8 waves
    scores_kernel<<<1024, 256, 0, stream>>>(key, mids, bias, out);
    // 64 rows -> 64 blocks
    softmax_norm_kernel<<<64, 256, 0, stream>>>(out);
}